// LoRAModulatedSchnetInteraction_23218593202761
// MI455X (gfx1250) — compile-verified
//
#include <hip/hip_runtime.h>
#include <hip/hip_bf16.h>
#include <stdint.h>

// ---------------------------------------------------------------------------
// LoRA-modulated SchNet interaction, fused for MI455X (gfx1250, wave32, WMMA)
//
// Roofline: heavy stage = 524288 pair-rows x (64->128->128->128) MLP
//   ~43 GFLOP fp32 ; f_ij read once = 134 MB (~6us @ 23.3 TB/s).
// Fusing the 3 layers + masked neighbor reduction avoids ~1.6 GB of HBM
// traffic for (B,A,A,128) intermediates -> compute-bound on fp32 WMMA.
// LoRA rank-4 folds exactly: h@W + (h@A)@B = h@(W + A@B)  (precomputed,
// stored transposed so WMMA B operands are contiguous b64 LDS loads).
// ---------------------------------------------------------------------------

typedef float v2f __attribute__((ext_vector_type(2)));
typedef float v8f __attribute__((ext_vector_type(8)));
typedef unsigned int u32x4 __attribute__((ext_vector_type(4)));
typedef unsigned int u32x8 __attribute__((ext_vector_type(8)));

// shifted softplus: log(0.5*e^v + 0.5) = softplus(v) - log(2), stable form
static __device__ __forceinline__ float sspf(float v) {
    return fmaxf(v, 0.0f) + __logf(1.0f + __expf(-fabsf(v))) - 0.69314718055994531f;
}

// D = A(16x4) * B(4x16) + C  on the native fp32 WMMA path
static __device__ __forceinline__ v8f wmma4(v2f a, v2f b, v8f c) {
    return __builtin_amdgcn_wmma_f32_16x16x4_f32(false, a, false, b, (short)0, c,
                                                 false, false);
}

static __device__ __forceinline__ unsigned int lds_off(const void* p) {
    // LDS aperture: LDS byte offset lives in addr[31:0]
    return (unsigned int)(uintptr_t)p;
}

// Tensor Data Mover: 2D tile (tileW x tileH elems, 4B each) global -> LDS.
// D# per CDNA5 ISA; groups 2/3 omitted (<=2D).
static __device__ __forceinline__ void tdm_load_2d(unsigned int ldsOff,
                                                   const void* gptr,
                                                   unsigned int tileW,
                                                   unsigned int tileH,
                                                   unsigned int strideElems) {
    unsigned long long ga = (unsigned long long)(uintptr_t)gptr;
    u32x4 g0;
    g0[0] = 1u;                                             // count=1, user D#
    g0[1] = ldsOff;                                         // lds_addr
    g0[2] = (unsigned int)ga;                               // global_addr[31:0]
    g0[3] = (unsigned int)((ga >> 32) & 0x01FFFFFFu) | (2u << 30); // [56:32] | type=2
    u32x8 g1;
    g1[0] = (2u << 16);                                     // wg_mask=0, data_size=4B
    g1[1] = (tileW & 0xFFFFu) << 16;                        // tensor_dim0[15:0]
    g1[2] = ((tileW >> 16) & 0xFFFFu) | ((tileH & 0xFFFFu) << 16); // dim0 hi | dim1 lo
    g1[3] = ((tileH >> 16) & 0xFFFFu) | ((tileW & 0xFFFFu) << 16); // dim1 hi | tile_dim0
    g1[4] = (tileH & 0xFFFFu);                              // tile_dim1 ; tile_dim2=0
    g1[5] = strideElems;                                    // tensor_dim0_stride[31:0]
    g1[6] = 0u;
    g1[7] = 0u;
    asm volatile("tensor_load_to_lds %0, %1" :: "s"(g0), "s"(g1) : "memory");
}

static __device__ __forceinline__ void tdm_wait() {
#if __has_builtin(__builtin_amdgcn_s_wait_tensorcnt)
    __builtin_amdgcn_s_wait_tensorcnt(0);
#else
    asm volatile("s_wait_tensorcnt 0x0" ::: "memory");
#endif
}

// ---------------------------------------------------------------------------
// Fold LoRA into effective weight, stored transposed:
//   dstT[n*K + k] = W[k,n] + sum_r A[b,k,r] * B[b,r,n]
// ---------------------------------------------------------------------------
__global__ void prep_wt_kernel(const float* __restrict__ W,
                               const float* __restrict__ A,
                               const float* __restrict__ B,
                               float* __restrict__ dstT, int K, int N, int b) {
    int idx = blockIdx.x * blockDim.x + threadIdx.x;
    if (idx >= K * N) return;
    int k = idx / N, n = idx % N;
    const float* Ab = A + (size_t)b * K * 4;
    const float* Bb = B + (size_t)b * 4 * N;
    float acc = W[(size_t)k * N + n];
#pragma unroll
    for (int r = 0; r < 4; ++r) acc += Ab[k * 4 + r] * Bb[r * N + n];
    dstT[(size_t)n * K + k] = acc;
}

// ---------------------------------------------------------------------------
// Generic 128->128 WMMA GEMM + bias + ssp (+ optional residual add).
// grid = 8 blocks x 256 threads (8 waves); each wave owns a 16-row M-tile.
// ---------------------------------------------------------------------------
__global__ __launch_bounds__(256) void dense_ssp_kernel(
    const float* __restrict__ in, const float* __restrict__ wtBase,
    int wtBatchStride, const float* __restrict__ bias,
    const float* __restrict__ residual, float* __restrict__ out) {
    __shared__ float sWT[128 * 128];

    const int tid = threadIdx.x;
    const int w = tid >> 5, lane = tid & 31, lm = lane & 15, lh = lane >> 4;
    const int r0 = blockIdx.x * 128;
    const int b = r0 >> 9;

    if (tid < 32) {
        tdm_load_2d(lds_off(sWT), wtBase + (size_t)b * wtBatchStride, 128u, 128u, 128u);
        tdm_wait();
    }
    __syncthreads();

    v8f acc[8];
#pragma unroll
    for (int nt = 0; nt < 8; ++nt) {
        float bf = bias[nt * 16 + lm];
        acc[nt] = (v8f){bf, bf, bf, bf, bf, bf, bf, bf};
    }

    const float* arow = in + (size_t)(r0 + 16 * w + lm) * 128;
#pragma unroll 4
    for (int kk = 0; kk < 32; ++kk) {
        v2f a = *(const v2f*)(arow + 4 * kk + 2 * lh);
#pragma unroll
        for (int nt = 0; nt < 8; ++nt) {
            v2f bv = *(const v2f*)(&sWT[(nt * 16 + lm) * 128 + 4 * kk + 2 * lh]);
            acc[nt] = wmma4(a, bv, acc[nt]);
        }
    }

#pragma unroll
    for (int nt = 0; nt < 8; ++nt) {
#pragma unroll
        for (int v = 0; v < 8; ++v) {
            int row = r0 + 16 * w + 8 * lh + v;
            int col = nt * 16 + lm;
            float val = sspf(acc[nt][v]);
            if (residual) val += residual[(size_t)row * 128 + col];
            out[(size_t)row * 128 + col] = val;
        }
    }
}

// ---------------------------------------------------------------------------
// Heavy fused kernel: one block per (b, i). 4 chunks of 128 j-rows.
// Per chunk: 3-layer WMMA MLP on f_ij rows (h in wave-private LDS between
// layers, TDM-staged weights), then masked/modulated reduction against x_i.
// ---------------------------------------------------------------------------
__global__ __launch_bounds__(256) void pair_kernel(
    const float* __restrict__ f_ij, const float* __restrict__ c_ij,
    const int* __restrict__ mask, const float* __restrict__ xi,
    const float* __restrict__ wtf0, const float* __restrict__ wtf1,
    const float* __restrict__ wtf2, const float* __restrict__ b_f0,
    const float* __restrict__ b_f1, const float* __restrict__ b_f2,
    float* __restrict__ y) {
    __shared__ float sWT[128 * 128];   // 64 KB: current layer's W_eff^T
    __shared__ float sH[128 * 128];    // 64 KB: h (wave-private rows) / xi tile
    __shared__ float sCoef[128];       // c_ij * mask for current chunk

    const int tid = threadIdx.x;
    const int w = tid >> 5, lane = tid & 31, lm = lane & 15, lh = lane >> 4;
    const int bi = blockIdx.x;
    const int b = bi >> 9;

    const float* WT0 = wtf0 + (size_t)b * (128 * 64);
    const float* WT1 = wtf1 + (size_t)b * (128 * 128);
    const float* WT2 = wtf2 + (size_t)b * (128 * 128);

    float bf0[8], bf1[8], bf2[8];
#pragma unroll
    for (int nt = 0; nt < 8; ++nt) {
        bf0[nt] = b_f0[nt * 16 + lm];
        bf1[nt] = b_f1[nt * 16 + lm];
        bf2[nt] = b_f2[nt * 16 + lm];
    }

    float yp[8];
#pragma unroll
    for (int nt = 0; nt < 8; ++nt) yp[nt] = 0.0f;

    const size_t pairBase = (size_t)bi * 512;  // index of (b,i,j=0)

    for (int j0 = 0; j0 < 512; j0 += 128) {
        // ---- stage layer-0 weights + chunk coefficients -------------------
        __syncthreads();
        if (tid < 32) {
            tdm_load_2d(lds_off(sWT), WT0, 64u, 128u, 64u);
            tdm_wait();
        }
        if (tid >= 128) {
            int j = j0 + tid - 128;
            sCoef[tid - 128] = c_ij[pairBase + j] * (float)mask[pairBase + j];
        }
        __syncthreads();

        // ---- layer 0: (64 -> 128), A from global f_ij ---------------------
        v8f acc[8];
#pragma unroll
        for (int nt = 0; nt < 8; ++nt)
            acc[nt] = (v8f){bf0[nt], bf0[nt], bf0[nt], bf0[nt],
                            bf0[nt], bf0[nt], bf0[nt], bf0[nt]};
        {
            const float* frow = f_ij + (pairBase + (size_t)j0 + 16 * w + lm) * 64;
#pragma unroll 4
            for (int kk = 0; kk < 16; ++kk) {
                v2f a = *(const v2f*)(frow + 4 * kk + 2 * lh);
#pragma unroll
                for (int nt = 0; nt < 8; ++nt) {
                    v2f bv = *(const v2f*)(&sWT[(nt * 16 + lm) * 64 + 4 * kk + 2 * lh]);
                    acc[nt] = wmma4(a, bv, acc[nt]);
                }
            }
        }
#pragma unroll
        for (int nt = 0; nt < 8; ++nt)
#pragma unroll
            for (int v = 0; v < 8; ++v)
                sH[(16 * w + 8 * lh + v) * 128 + nt * 16 + lm] = sspf(acc[nt][v]);

        // ---- layer 1: (128 -> 128), A from LDS h --------------------------
        __syncthreads();
        if (tid < 32) {
            tdm_load_2d(lds_off(sWT), WT1, 128u, 128u, 128u);
            tdm_wait();
        }
        __syncthreads();
#pragma unroll
        for (int nt = 0; nt < 8; ++nt)
            acc[nt] = (v8f){bf1[nt], bf1[nt], bf1[nt], bf1[nt],
                            bf1[nt], bf1[nt], bf1[nt], bf1[nt]};
#pragma unroll 4
        for (int kk = 0; kk < 32; ++kk) {
            v2f a = *(const v2f*)(&sH[(16 * w + lm) * 128 + 4 * kk + 2 * lh]);
#pragma unroll
            for (int nt = 0; nt < 8; ++nt) {
                v2f bv = *(const v2f*)(&sWT[(nt * 16 + lm) * 128 + 4 * kk + 2 * lh]);
                acc[nt] = wmma4(a, bv, acc[nt]);
            }
        }
#pragma unroll
        for (int nt = 0; nt < 8; ++nt)
#pragma unroll
            for (int v = 0; v < 8; ++v)
                sH[(16 * w + 8 * lh + v) * 128 + nt * 16 + lm] = sspf(acc[nt][v]);

        // ---- layer 2: (128 -> 128), result stays in registers -------------
        __syncthreads();
        if (tid < 32) {
            tdm_load_2d(lds_off(sWT), WT2, 128u, 128u, 128u);
            tdm_wait();
        }
        __syncthreads();
#pragma unroll
        for (int nt = 0; nt < 8; ++nt)
            acc[nt] = (v8f){bf2[nt], bf2[nt], bf2[nt], bf2[nt],
                            bf2[nt], bf2[nt], bf2[nt], bf2[nt]};
#pragma unroll 4
        for (int kk = 0; kk < 32; ++kk) {
            v2f a = *(const v2f*)(&sH[(16 * w + lm) * 128 + 4 * kk + 2 * lh]);
#pragma unroll
            for (int nt = 0; nt < 8; ++nt) {
                v2f bv = *(const v2f*)(&sWT[(nt * 16 + lm) * 128 + 4 * kk + 2 * lh]);
                acc[nt] = wmma4(a, bv, acc[nt]);
            }
        }

        // ---- masked, c_ij-modulated reduction against x_i -----------------
        __syncthreads();
        if (tid < 32) {
            tdm_load_2d(lds_off(sH), xi + ((size_t)b * 512 + j0) * 128, 128u, 128u, 128u);
            tdm_wait();
        }
        __syncthreads();
#pragma unroll
        for (int nt = 0; nt < 8; ++nt)
#pragma unroll
            for (int v = 0; v < 8; ++v) {
                int row = 16 * w + 8 * lh + v;
                float h2 = sspf(acc[nt][v]);
                yp[nt] += sCoef[row] * sH[row * 128 + nt * 16 + lm] * h2;
            }
    }

    // ---- cross-wave reduction of y partials -------------------------------
    __syncthreads();
#pragma unroll
    for (int nt = 0; nt < 8; ++nt)
        sWT[(w * 2 + lh) * 128 + nt * 16 + lm] = yp[nt];
    __syncthreads();
    if (tid < 128) {
        float s = 0.0f;
#pragma unroll
        for (int k = 0; k < 16; ++k) s += sWT[k * 128 + tid];
        y[(size_t)bi * 128 + tid] = s;
    }
}

// ---------------------------------------------------------------------------
// Host launcher
// ---------------------------------------------------------------------------
extern "C" void kernel_launch(void* const* d_in, const int* in_sizes, int n_in,
                              void* d_out, int out_size, void* d_ws, size_t ws_size,
                              hipStream_t stream) {
    const float* x    = (const float*)d_in[0];
    const float* f_ij = (const float*)d_in[1];
    const float* c_ij = (const float*)d_in[2];
    const int*   mask = (const int*)d_in[3];

    const float* W[6]  = {(const float*)d_in[4],  (const float*)d_in[8],
                          (const float*)d_in[12], (const float*)d_in[16],
                          (const float*)d_in[20], (const float*)d_in[24]};
    const float* bs[6] = {(const float*)d_in[5],  (const float*)d_in[9],
                          (const float*)d_in[13], (const float*)d_in[17],
                          (const float*)d_in[21], (const float*)d_in[25]};
    const float* Aa[6] = {(const float*)d_in[6],  (const float*)d_in[10],
                          (const float*)d_in[14], (const float*)d_in[18],
                          (const float*)d_in[22], (const float*)d_in[26]};
    const float* Bb[6] = {(const float*)d_in[7],  (const float*)d_in[11],
                          (const float*)d_in[15], (const float*)d_in[19],
                          (const float*)d_in[23], (const float*)d_in[27]};
    const int Kdim[6] = {128, 64, 128, 128, 128, 128};

    float* ws    = (float*)d_ws;
    float* wt_bc = ws;            // 2 * 128*128
    float* wt_f0 = ws + 32768;    // 2 * 128*64
    float* wt_f1 = ws + 49152;    // 2 * 128*128
    float* wt_f2 = ws + 81920;
    float* wt_a0 = ws + 114688;
    float* wt_a1 = ws + 147456;
    float* xi    = ws + 180224;   // 2*512*128
    float* yv    = ws + 311296;   // 2*512*128
    float* t0    = ws + 442368;   // 2*512*128
    float* wt[6] = {wt_bc, wt_f0, wt_f1, wt_f2, wt_a0, wt_a1};

    // 1) Fold LoRA into per-batch transposed effective weights
    for (int l = 0; l < 6; ++l) {
        for (int b = 0; b < 2; ++b) {
            int K = Kdim[l], N = 128;
            int nthreads = K * N;
            prep_wt_kernel<<<(nthreads + 255) / 256, 256, 0, stream>>>(
                W[l], Aa[l], Bb[l], wt[l] + (size_t)b * N * K, K, N, b);
        }
    }

    // 2) x_i = ssp(x @ W_bc_eff + b_bc)
    dense_ssp_kernel<<<8, 256, 0, stream>>>(x, wt_bc, 128 * 128, bs[0], nullptr, xi);

    // 3) Fused pairwise MLP + masked neighbor reduction -> y
    pair_kernel<<<1024, 256, 0, stream>>>(f_ij, c_ij, mask, xi, wt_f0, wt_f1, wt_f2,
                                          bs[1], bs[2], bs[3], yv);

    // 4) y1 = ssp(y @ Wa0_eff + b_a0) ; 5) out = x + ssp(y1 @ Wa1_eff + b_a1)
    dense_ssp_kernel<<<8, 256, 0, stream>>>(yv, wt_a0, 128 * 128, bs[4], nullptr, t0);
    dense_ssp_kernel<<<8, 256, 0, stream>>>(t0, wt_a1, 128 * 128, bs[5], x,
                                            (float*)d_out);
}